// ChamferDistanceL1_7473243095507
// MI455X (gfx1250) — compile-verified
//
#include <hip/hip_runtime.h>

typedef float v2f __attribute__((ext_vector_type(2)));
typedef float v8f __attribute__((ext_vector_type(8)));

#define EPS_SQ 1e-12f

__global__ void chamfer_init_kernel(float* out) { out[0] = 0.0f; }

// One block = 256 threads = 8 wave32. Each wave owns TWO 16-row tiles of X
// (32 rows) and sweeps all of Y (staged in LDS) in 16-column tiles.
// Full-K formulation: A[m] = (px,py,pz,1), B[n] = (-2qx,-2qy,-2qz,|q|^2),
// C[m] = |x_m|^2 (loop-invariant). One V_WMMA_F32_16X16X4_F32 then yields
// the complete squared distance d(m,n) = |x|^2 + |y|^2 - 2 x.y directly;
// the loop body is just load-B, 2 WMMAs, and running mins.
__global__ __launch_bounds__(256) void chamfer_dir_kernel(
    const float* __restrict__ X,   // [B, N, 3] "query" points (row set)
    const float* __restrict__ Y,   // [B, N, 3] "target" points (column set)
    float* __restrict__ out,       // scalar accumulator
    int N, float scale)
{
    extern __shared__ float4 yStage[];   // N entries: (-2qx, -2qy, -2qz, |q|^2)

    const int  tid  = threadIdx.x;
    const int  b    = blockIdx.y;
    const long base = (long)b * N * 3;

    // ---- Stage all of Y for this batch into LDS (8192 pts -> 128KB)
    for (int i = tid; i < N; i += 256) {
        float qx = Y[base + 3 * i + 0];
        float qy = Y[base + 3 * i + 1];
        float qz = Y[base + 3 * i + 2];
        yStage[i] = make_float4(-2.0f * qx, -2.0f * qy, -2.0f * qz,
                                qx * qx + qy * qy + qz * qz);
    }
    __syncthreads();

    const int  lane = tid & 31;
    const int  wave = tid >> 5;
    const int  l15  = lane & 15;
    const bool hi   = lane >= 16;

    // ---- Load this wave's 2x16 X rows; build loop-invariant A operands.
    // A 16x4 f32 layout: lanes 0-15 hold K=0,1 ; lanes 16-31 hold K=2,3.
    const int row0 = blockIdx.x * 256 + wave * 32;   // first tile rows
    const int row1 = row0 + 16;                      // second tile rows

    const int p0 = row0 + l15;
    float p0x = X[base + 3 * p0 + 0];
    float p0y = X[base + 3 * p0 + 1];
    float p0z = X[base + 3 * p0 + 2];
    float x2a = p0x * p0x + p0y * p0y + p0z * p0z;

    const int p1 = row1 + l15;
    float p1x = X[base + 3 * p1 + 0];
    float p1y = X[base + 3 * p1 + 1];
    float p1z = X[base + 3 * p1 + 2];
    float x2b = p1x * p1x + p1y * p1y + p1z * p1z;

    v2f a0, a1;
    a0.x = hi ? p0z : p0x;
    a0.y = hi ? 1.0f : p0y;        // K=3 slot carries the 1 multiplying |y|^2
    a1.x = hi ? p1z : p1x;
    a1.y = hi ? 1.0f : p1y;

    // Loop-invariant C operands: |x_m|^2 for the row each C/D slot r maps to
    // (m = r + 8*hi). Gather via shuffles once.
    v8f ci0, ci1;
#pragma unroll
    for (int r = 0; r < 8; ++r) {
        ci0[r] = __shfl(x2a, r + (hi ? 8 : 0), 32);
        ci1[r] = __shfl(x2b, r + (hi ? 8 : 0), 32);
    }

    float rm0[8], rm1[8];
#pragma unroll
    for (int r = 0; r < 8; ++r) { rm0[r] = 3.4e38f; rm1[r] = 3.4e38f; }

    // ---- Sweep all column tiles of Y; each staged B feeds two WMMAs.
    const int ntiles = N >> 4;
    for (int t = 0; t < ntiles; ++t) {
        float4 q = yStage[t * 16 + l15];
        // B 4x16 f32 layout mirrors A: lanes 0-15 K=0,1 ; lanes 16-31 K=2,3.
        v2f bv;
        bv.x = hi ? q.z : q.x;     // -2*q coords staged in LDS
        bv.y = hi ? q.w : q.y;     // K=3 slot = |q|^2

        // D = A x B + C  ==  full squared distance tile.
        v8f d0 = __builtin_amdgcn_wmma_f32_16x16x4_f32(
                     false, a0, false, bv, (short)0, ci0, false, false);
        v8f d1 = __builtin_amdgcn_wmma_f32_16x16x4_f32(
                     false, a1, false, bv, (short)0, ci1, false, false);

#pragma unroll
        for (int r = 0; r < 8; ++r) {
            rm0[r] = fminf(rm0[r], d0[r]);
            rm1[r] = fminf(rm1[r], d1[r]);
        }
    }

    // ---- Column reduction: min across the 16 lanes of each half-wave.
#pragma unroll
    for (int off = 1; off < 16; off <<= 1) {
#pragma unroll
        for (int r = 0; r < 8; ++r) {
            rm0[r] = fminf(rm0[r], __shfl_xor(rm0[r], off, 32));
            rm1[r] = fminf(rm1[r], __shfl_xor(rm1[r], off, 32));
        }
    }

    // Lane 0 holds rows 0-7 / 16-23, lane 16 holds rows 8-15 / 24-31.
    if (l15 == 0) {
        float s = 0.0f;
#pragma unroll
        for (int r = 0; r < 8; ++r) {
            s += sqrtf(fmaxf(rm0[r], EPS_SQ));
            s += sqrtf(fmaxf(rm1[r], EPS_SQ));
        }
        atomicAdd(out, s * scale);
    }
}

extern "C" void kernel_launch(void* const* d_in, const int* in_sizes, int n_in,
                              void* d_out, int out_size, void* d_ws, size_t ws_size,
                              hipStream_t stream) {
    const float* xyz1 = (const float*)d_in[0];
    const float* xyz2 = (const float*)d_in[1];
    float* out = (float*)d_out;

    const int N = 8192;                       // points per cloud (reference)
    const int B = in_sizes[0] / (3 * N);      // batches

    // Final scalar = sum of all sqrt'd per-point min distances (both
    // directions) / (2 * N * B).
    const float scale = 1.0f / (2.0f * (float)N * (float)B);

    chamfer_init_kernel<<<1, 1, 0, stream>>>(out);

    dim3 grid(N / 256, B);                    // 32 x 8 workgroups
    size_t ldsBytes = (size_t)N * sizeof(float4);   // 128 KB < 320 KB WGP LDS

    // Direction 1: for each xyz1 point, nearest xyz2 point.
    chamfer_dir_kernel<<<grid, 256, ldsBytes, stream>>>(xyz1, xyz2, out, N, scale);
    // Direction 2: swapped roles.
    chamfer_dir_kernel<<<grid, 256, ldsBytes, stream>>>(xyz2, xyz1, out, N, scale);
}